// ControlledNODE_59064390255152
// MI455X (gfx1250) — compile-verified
//
#include <hip/hip_runtime.h>
#include <stddef.h>

// ---------------------------------------------------------------------------
// Controlled neural ODE scan, MI455X (gfx1250, wave32).
// Phase 1: P = U @ W1[16:48] + b1        -- fp32 WMMA GEMM (parallel)
// Phase 2: sequential Euler scan          -- single wave32, LDS-resident weights
// Phase 3: [d,t,l] = H @ [wd|wt|wl] + b  -- fp32 WMMA GEMM (parallel)
// ---------------------------------------------------------------------------

typedef __attribute__((ext_vector_type(2))) float v2f;
typedef __attribute__((ext_vector_type(8))) float v8f;

__device__ __forceinline__ float fast_tanhf(float x) {
  // tanh(x) = sign(x) * (1 - 2/(exp(2|x|)+1)); stable for large |x| (exp->inf -> 1)
  float ax = __builtin_fabsf(x);
  float e  = __expf(2.0f * ax);                      // v_exp_f32 path
  float r  = 1.0f - 2.0f * __builtin_amdgcn_rcpf(e + 1.0f);
  return __builtin_copysignf(r, x);
}

// ---------------------------------------------------------------------------
// Phase 1: P[t][c] = b1[c] + sum_{k<32} U[t][k] * W1[16+k][c]
// One wave per 16-row tile. fp32 WMMA 16x16x4:
//   A 16x4 : lanes 0-15 rows, v0=K0/K2(lane-hi), v1=K1/K3
//   B 4x16 : lanes 0-15 cols, mirrored K split
//   C/D    : VGPR v -> row v (lanes 0-15) / row v+8 (lanes 16-31), col = lane&15
// ---------------------------------------------------------------------------
__global__ __launch_bounds__(256) void node_p1(const float* __restrict__ U,
                                               const float* __restrict__ W1,
                                               const float* __restrict__ b1,
                                               float* __restrict__ P,
                                               int ntiles) {
  int lane = threadIdx.x & 31;
  int wave = threadIdx.x >> 5;
  int tile = blockIdx.x * 8 + wave;
  if (tile >= ntiles) return;                       // wave-uniform guard
  int r  = lane & 15;                               // row (A) / col (B,D)
  int kh = lane >> 4;                               // K-half select

  const float* urow = U + (size_t)(tile * 16 + r) * 32;

  v8f acc[4];
#pragma unroll
  for (int nt = 0; nt < 4; ++nt) {
    float bv = b1[nt * 16 + r];
#pragma unroll
    for (int v = 0; v < 8; ++v) acc[nt][v] = bv;    // bias broadcast into C
  }

#pragma unroll
  for (int s = 0; s < 8; ++s) {                     // K = 32 in steps of 4
    int k = 4 * s + 2 * kh;
    v2f a;
    a.x = urow[k];
    a.y = urow[k + 1];
#pragma unroll
    for (int nt = 0; nt < 4; ++nt) {
      v2f bm;
      bm.x = W1[(size_t)(16 + k) * 64 + nt * 16 + r];
      bm.y = W1[(size_t)(17 + k) * 64 + nt * 16 + r];
      acc[nt] = __builtin_amdgcn_wmma_f32_16x16x4_f32(
          false, a, false, bm, (short)0, acc[nt], false, false);
    }
  }

#pragma unroll
  for (int nt = 0; nt < 4; ++nt)
#pragma unroll
    for (int v = 0; v < 8; ++v)
      P[(size_t)(tile * 16 + v + 8 * kh) * 64 + nt * 16 + r] = acc[nt][v];
}

// ---------------------------------------------------------------------------
// Phase 3: readouts from stored pre-update states H[T][16].
// out[0:T)=d, [T:2T)=t, [2T:3T)=l.  Wr is 16x16 with cols 0..2 = wd,wt,wl.
// ---------------------------------------------------------------------------
__global__ __launch_bounds__(256) void node_p3(const float* __restrict__ H,
                                               const float* __restrict__ wd,
                                               const float* __restrict__ bd,
                                               const float* __restrict__ wt,
                                               const float* __restrict__ bt,
                                               const float* __restrict__ wl,
                                               const float* __restrict__ bl,
                                               float* __restrict__ out,
                                               int ntiles, size_t T) {
  __shared__ __align__(16) float sWr[16 * 16];
  int tid = threadIdx.x;
  sWr[tid] = 0.0f;
  __syncthreads();
  if (tid < 16) {
    sWr[tid * 16 + 0] = wd[tid];
    sWr[tid * 16 + 1] = wt[tid];
    sWr[tid * 16 + 2] = wl[tid];
  }
  __syncthreads();

  int lane = tid & 31;
  int wave = tid >> 5;
  int tile = blockIdx.x * 8 + wave;
  if (tile >= ntiles) return;
  int r  = lane & 15;
  int kh = lane >> 4;

  float bv = (r == 0) ? bd[0] : (r == 1) ? bt[0] : (r == 2) ? bl[0] : 0.0f;
  v8f acc;
#pragma unroll
  for (int v = 0; v < 8; ++v) acc[v] = bv;

  const float* hrow = H + (size_t)(tile * 16 + r) * 16;
#pragma unroll
  for (int s = 0; s < 4; ++s) {                     // K = 16 in steps of 4
    int k = 4 * s + 2 * kh;
    v2f a;
    a.x = hrow[k];
    a.y = hrow[k + 1];
    v2f bm;
    bm.x = sWr[k * 16 + r];
    bm.y = sWr[(k + 1) * 16 + r];
    acc = __builtin_amdgcn_wmma_f32_16x16x4_f32(
        false, a, false, bm, (short)0, acc, false, false);
  }

  if (r < 3) {
#pragma unroll
    for (int v = 0; v < 8; ++v)
      out[(size_t)r * T + (size_t)(tile * 16 + v + 8 * kh)] = acc[v];
  }
}

// ---------------------------------------------------------------------------
// Phase 2: the sequential scan. ONE wave32 (no barriers needed: same-wave DS
// ops are in-order). Lane j owns columns j and j+32 of each 64-wide layer.
// Weights in LDS, column-major, 16B-aligned strides (b128 loads, <=2-way bank
// conflict). P stream software-pipelined + global_prefetch.
// ---------------------------------------------------------------------------
template <bool USE_P, bool STORE_H>
__global__ __launch_bounds__(32) void node_scan(
    const float* __restrict__ U, const float* __restrict__ h0,
    const float* __restrict__ W1, const float* __restrict__ b1,
    const float* __restrict__ W2, const float* __restrict__ b2,
    const float* __restrict__ W3, const float* __restrict__ b3,
    const float* __restrict__ wd, const float* __restrict__ bd,
    const float* __restrict__ wt, const float* __restrict__ bt,
    const float* __restrict__ wl, const float* __restrict__ bl,
    const float* __restrict__ P, float* __restrict__ H,
    float* __restrict__ out, int T) {
  __shared__ __align__(16) float sW1h[64 * 20];   // W1h^T  [col][k0..15], stride 20
  __shared__ __align__(16) float sW1u[64 * 36];   // W1u^T  [col][k0..31], stride 36
  __shared__ __align__(16) float sW2[64 * 68];    // W2^T   [col][k0..63], stride 68
  __shared__ __align__(16) float sW3[16 * 68];    // W3^T   [col][k0..63], stride 68
  __shared__ __align__(16) float sZ1[64];
  __shared__ __align__(16) float sZ2[64];
  __shared__ __align__(16) float sH[16];

  const int j = threadIdx.x;                      // 0..31

  // ---- stage weights (transposed) into LDS ----
  for (int idx = j; idx < 16 * 64; idx += 32) {
    int k = idx >> 6, c = idx & 63;
    sW1h[c * 20 + k] = W1[(size_t)k * 64 + c];
  }
  if (!USE_P) {
    for (int idx = j; idx < 32 * 64; idx += 32) {
      int k = idx >> 6, c = idx & 63;
      sW1u[c * 36 + k] = W1[(size_t)(16 + k) * 64 + c];
    }
  }
  for (int idx = j; idx < 64 * 64; idx += 32) {
    int k = idx >> 6, c = idx & 63;
    sW2[c * 68 + k] = W2[(size_t)k * 64 + c];
  }
  for (int idx = j; idx < 64 * 16; idx += 32) {
    int k = idx >> 4, c = idx & 15;
    sW3[c * 68 + k] = W3[(size_t)k * 16 + c];
  }
  if (j < 16) sH[j] = h0[j];
  __syncthreads();

  const float dt  = (float)(5.0 / 60.0);
  const float b2a = b2[j], b2b = b2[j + 32];
  float b1a = 0.0f, b1b = 0.0f;
  if (!USE_P) { b1a = b1[j]; b1b = b1[j + 32]; }
  const float b3i = (j < 16) ? b3[j] : 0.0f;
  float hi        = (j < 16) ? h0[j] : 0.0f;

  float rwd[16], rwt[16], rwl[16];
  float bd0 = 0.0f, bt0 = 0.0f, bl0 = 0.0f;
  if constexpr (!STORE_H) {
#pragma unroll
    for (int k = 0; k < 16; ++k) { rwd[k] = wd[k]; rwt[k] = wt[k]; rwl[k] = wl[k]; }
    bd0 = bd[0]; bt0 = bt[0]; bl0 = bl[0];
  }

  float p0 = 0.0f, p1 = 0.0f;
  if (USE_P) { p0 = P[j]; p1 = P[j + 32]; }

  for (int t = 0; t < T; ++t) {
    // ---- issue next-step P loads + prefetch early (latency hiding) ----
    float pn0 = 0.0f, pn1 = 0.0f;
    if (USE_P) {
      int tn = (t + 1 < T) ? t + 1 : t;
      pn0 = P[(size_t)tn * 64 + j];
      pn1 = P[(size_t)tn * 64 + j + 32];
      int tp = (t + 8 < T) ? t + 8 : T - 1;
      __builtin_prefetch(P + (size_t)tp * 64 + j, 0, 1);   // global_prefetch_b8
    }

    // ---- broadcast h from LDS (pre-update state of step t) ----
    float4 h0v = *(const float4*)&sH[0];
    float4 h1v = *(const float4*)&sH[4];
    float4 h2v = *(const float4*)&sH[8];
    float4 h3v = *(const float4*)&sH[12];
    float hk[16] = {h0v.x, h0v.y, h0v.z, h0v.w, h1v.x, h1v.y, h1v.z, h1v.w,
                    h2v.x, h2v.y, h2v.z, h2v.w, h3v.x, h3v.y, h3v.z, h3v.w};

    if constexpr (STORE_H) {
      if (j < 16) H[(size_t)t * 16 + j] = hi;     // fire-and-forget store
    } else {
      float dv = bd0, tv = bt0, lv = bl0;
#pragma unroll
      for (int k = 0; k < 16; ++k) {
        dv += hk[k] * rwd[k];
        tv += hk[k] * rwt[k];
        lv += hk[k] * rwl[k];
      }
      if (j < 3)
        out[(size_t)j * (size_t)T + t] = (j == 0) ? dv : ((j == 1) ? tv : lv);
    }

    // ---- layer 1: z1 = tanh(P[t] + h @ W1h)  (cols j, j+32) ----
    float a0 = USE_P ? p0 : b1a;
    float a1 = USE_P ? p1 : b1b;
    {
      const float4* w0 = (const float4*)&sW1h[j * 20];
      const float4* w1 = (const float4*)&sW1h[(j + 32) * 20];
#pragma unroll
      for (int q = 0; q < 4; ++q) {
        float4 wa = w0[q], wb = w1[q];
        a0 += hk[4*q+0]*wa.x + hk[4*q+1]*wa.y + hk[4*q+2]*wa.z + hk[4*q+3]*wa.w;
        a1 += hk[4*q+0]*wb.x + hk[4*q+1]*wb.y + hk[4*q+2]*wb.z + hk[4*q+3]*wb.w;
      }
    }
    if constexpr (!USE_P) {
      const float4* uv  = (const float4*)(U + (size_t)t * 32);
      const float4* wu0 = (const float4*)&sW1u[j * 36];
      const float4* wu1 = (const float4*)&sW1u[(j + 32) * 36];
#pragma unroll
      for (int q = 0; q < 8; ++q) {
        float4 u4 = uv[q];
        float4 wa = wu0[q], wb = wu1[q];
        a0 += u4.x*wa.x + u4.y*wa.y + u4.z*wa.z + u4.w*wa.w;
        a1 += u4.x*wb.x + u4.y*wb.y + u4.z*wb.z + u4.w*wb.w;
      }
    }
    sZ1[j]      = fast_tanhf(a0);
    sZ1[j + 32] = fast_tanhf(a1);

    // ---- layer 2: z2 = tanh(z1 @ W2 + b2) ----
    float c0 = b2a, c1 = b2b;
    {
      const float4* zp = (const float4*)&sZ1[0];
      const float4* w0 = (const float4*)&sW2[j * 68];
      const float4* w1 = (const float4*)&sW2[(j + 32) * 68];
#pragma unroll
      for (int q = 0; q < 16; ++q) {
        float4 z = zp[q];                          // broadcast read
        float4 wa = w0[q], wb = w1[q];
        c0 += z.x*wa.x + z.y*wa.y + z.z*wa.z + z.w*wa.w;
        c1 += z.x*wb.x + z.y*wb.y + z.z*wb.z + z.w*wb.w;
      }
    }
    sZ2[j]      = fast_tanhf(c0);
    sZ2[j + 32] = fast_tanhf(c1);

    // ---- layer 3 + Euler update: h += dt * (z2 @ W3 + b3) ----
    if (j < 16) {
      float acc = b3i;
      const float4* zp = (const float4*)&sZ2[0];
      const float4* w3 = (const float4*)&sW3[j * 68];
#pragma unroll
      for (int q = 0; q < 16; ++q) {
        float4 z = zp[q];
        float4 w = w3[q];
        acc += z.x*w.x + z.y*w.y + z.z*w.z + z.w*w.w;
      }
      hi += acc * dt;
      sH[j] = hi;                                  // same-wave DS: in-order
    }

    if (USE_P) { p0 = pn0; p1 = pn1; }
  }

  if (j < 16) out[(size_t)3 * (size_t)T + j] = hi; // h_final
}

// ---------------------------------------------------------------------------
// Host launcher
// ---------------------------------------------------------------------------
extern "C" void kernel_launch(void* const* d_in, const int* in_sizes, int n_in,
                              void* d_out, int out_size, void* d_ws, size_t ws_size,
                              hipStream_t stream) {
  const float* U  = (const float*)d_in[0];
  const float* h0 = (const float*)d_in[1];
  const float* W1 = (const float*)d_in[2];
  const float* b1 = (const float*)d_in[3];
  const float* W2 = (const float*)d_in[4];
  const float* b2 = (const float*)d_in[5];
  const float* W3 = (const float*)d_in[6];
  const float* b3 = (const float*)d_in[7];
  const float* wd = (const float*)d_in[8];
  const float* bd = (const float*)d_in[9];
  const float* wt = (const float*)d_in[10];
  const float* bt = (const float*)d_in[11];
  const float* wl = (const float*)d_in[12];
  const float* bl = (const float*)d_in[13];
  float* out = (float*)d_out;

  const int T = in_sizes[0] / 32;                 // 262144
  const size_t needP = (size_t)T * 64 * sizeof(float);   // 64 MB
  const size_t needH = (size_t)T * 16 * sizeof(float);   // 16 MB

  float* P = nullptr;
  float* H = nullptr;
  char* ws = (char*)d_ws;
  if (ws_size >= needP + needH) { P = (float*)ws; H = (float*)(ws + needP); }
  else if (ws_size >= needP)    { P = (float*)ws; }
  else if (ws_size >= needH)    { H = (float*)ws; }

  const int ntiles = T / 16;
  const int nblk   = (ntiles + 7) / 8;

  if (P) node_p1<<<nblk, 256, 0, stream>>>(U, W1, b1, P, ntiles);

  if (P && H)
    node_scan<true, true><<<1, 32, 0, stream>>>(U, h0, W1, b1, W2, b2, W3, b3,
                                                wd, bd, wt, bt, wl, bl, P, H, out, T);
  else if (P)
    node_scan<true, false><<<1, 32, 0, stream>>>(U, h0, W1, b1, W2, b2, W3, b3,
                                                 wd, bd, wt, bt, wl, bl, P, H, out, T);
  else if (H)
    node_scan<false, true><<<1, 32, 0, stream>>>(U, h0, W1, b1, W2, b2, W3, b3,
                                                 wd, bd, wt, bt, wl, bl, P, H, out, T);
  else
    node_scan<false, false><<<1, 32, 0, stream>>>(U, h0, W1, b1, W2, b2, W3, b3,
                                                  wd, bd, wt, bt, wl, bl, P, H, out, T);

  if (H) node_p3<<<nblk, 256, 0, stream>>>(H, wd, bd, wt, bt, wl, bl, out, ntiles, (size_t)T);
}